// SoftLength_23699629540183
// MI455X (gfx1250) — compile-verified
//
#include <hip/hip_runtime.h>
#include <math.h>

// Problem constants (from the reference): x is [L, B] row-major fp32.
#define L_DIM   8192
#define B_DIM   4096
#define C_CHUNKS  64                 // chunks along L
#define LC      (L_DIM / C_CHUNKS)   // 128 rows per chunk
#define B4      (B_DIM / 4)          // 1024 float4 column-groups per row
#define NEG_MAX (-3.402823466e38f)

__device__ __forceinline__ float sigmoidf_(float x) {
    // sigmoid(x) = 1 / (1 + e^-x); v_exp_f32 + accurate divide
    return 1.0f / (1.0f + __expf(-x));
}

// ---------------------------------------------------------------------------
// K1: per-chunk max of raw x (sigmoid is monotone, so max commutes with it).
// Grid: (B4/256, C_CHUNKS). blockIdx.y = chunk k, thread x = float4 column c.
// Pure streaming pass: 128 MiB at HBM rate; also warms the 192 MB L2 so the
// selective re-read in K3 hits L2.
// ---------------------------------------------------------------------------
__global__ void sl_chunkmax_kernel(const float4* __restrict__ x,
                                   float4* __restrict__ cmax) {
    const unsigned k = blockIdx.y;
    const unsigned c = blockIdx.x * blockDim.x + threadIdx.x;   // 0 .. B4-1
    const float4* p = x + (size_t)k * LC * B4 + c;

    float m0 = NEG_MAX, m1 = NEG_MAX, m2 = NEG_MAX, m3 = NEG_MAX;
#pragma unroll 8
    for (unsigned i = 0; i < LC; ++i) {
        if ((i & 3u) == 0u) {
            // gfx1250 prefetch path (global_prefetch_b8); speculative, OOB-safe,
            // one per 4-load group, 32 rows (512 KiB of stream) ahead.
            __builtin_prefetch((const void*)(p + (size_t)(i + 32) * B4), 0, 0);
        }
        float4 v = p[(size_t)i * B4];
        m0 = fmaxf(m0, v.x);
        m1 = fmaxf(m1, v.y);
        m2 = fmaxf(m2, v.z);
        m3 = fmaxf(m3, v.w);
    }
    cmax[(size_t)k * B4 + c] = make_float4(m0, m1, m2, m3);
}

// ---------------------------------------------------------------------------
// K2: inclusive suffix max over chunks: smax[k] = max(cmax[k..C-1]) per column.
// Only B4 = 1024 threads; 1 MiB of L2-resident traffic.
// ---------------------------------------------------------------------------
__global__ void sl_suffixmax_kernel(const float4* __restrict__ cmax,
                                    float4* __restrict__ smax) {
    const unsigned c = blockIdx.x * blockDim.x + threadIdx.x;   // 0 .. B4-1
    float m0 = NEG_MAX, m1 = NEG_MAX, m2 = NEG_MAX, m3 = NEG_MAX;
    for (int k = C_CHUNKS - 1; k >= 0; --k) {
        float4 v = cmax[(size_t)k * B4 + c];
        m0 = fmaxf(m0, v.x);
        m1 = fmaxf(m1, v.y);
        m2 = fmaxf(m2, v.z);
        m3 = fmaxf(m3, v.w);
        smax[(size_t)k * B4 + c] = make_float4(m0, m1, m2, m3);
    }
}

// ---------------------------------------------------------------------------
// K3: per-chunk partial sums of the reverse-cummax sigmoid.
//  - If the chunk's max is dominated by the suffix max of later chunks, the
//    running max is constant across the chunk: partial = LC * sigmoid(M_in),
//    no data re-read (~93% of chunks for i.i.d. data; expected re-read
//    fraction is H_64/64 ~= 7%, and those rows are L2-resident after K1).
//  - Otherwise scan the chunk backwards maintaining the running max.
// Both paths compute the exact same mathematical result.
// ---------------------------------------------------------------------------
__global__ void sl_scan_kernel(const float4* __restrict__ x,
                               const float4* __restrict__ cmax,
                               const float4* __restrict__ smax,
                               float4* __restrict__ partial) {
    const unsigned k = blockIdx.y;
    const unsigned c = blockIdx.x * blockDim.x + threadIdx.x;   // 0 .. B4-1

    float4 Min;
    if (k == C_CHUNKS - 1) {
        Min = make_float4(NEG_MAX, NEG_MAX, NEG_MAX, NEG_MAX);
    } else {
        Min = smax[(size_t)(k + 1) * B4 + c];
    }
    const float4 cm = cmax[(size_t)k * B4 + c];

    if (cm.x <= Min.x && cm.y <= Min.y && cm.z <= Min.z && cm.w <= Min.w) {
        // Incoming tail max dominates the whole chunk: closed form, no re-read.
        float4 pr;
        pr.x = (float)LC * sigmoidf_(Min.x);
        pr.y = (float)LC * sigmoidf_(Min.y);
        pr.z = (float)LC * sigmoidf_(Min.z);
        pr.w = (float)LC * sigmoidf_(Min.w);
        partial[(size_t)k * B4 + c] = pr;
        return;
    }

    float m0 = Min.x, m1 = Min.y, m2 = Min.z, m3 = Min.w;
    float s0 = 0.0f, s1 = 0.0f, s2 = 0.0f, s3 = 0.0f;
    const float4* p = x + (size_t)k * LC * B4 + c;
#pragma unroll 4
    for (int i = LC - 1; i >= 0; --i) {
        float4 v = p[(size_t)i * B4];
        m0 = fmaxf(m0, v.x);  s0 += sigmoidf_(m0);
        m1 = fmaxf(m1, v.y);  s1 += sigmoidf_(m1);
        m2 = fmaxf(m2, v.z);  s2 += sigmoidf_(m2);
        m3 = fmaxf(m3, v.w);  s3 += sigmoidf_(m3);
    }
    partial[(size_t)k * B4 + c] = make_float4(s0, s1, s2, s3);
}

// ---------------------------------------------------------------------------
// K4: deterministic reduction of the C_CHUNKS partials per column -> out[B].
// Fixed summation order => bitwise-stable output across replays (no atomics).
// ---------------------------------------------------------------------------
__global__ void sl_reduce_kernel(const float4* __restrict__ partial,
                                 float4* __restrict__ out) {
    const unsigned c = blockIdx.x * blockDim.x + threadIdx.x;   // 0 .. B4-1
    float s0 = 0.0f, s1 = 0.0f, s2 = 0.0f, s3 = 0.0f;
#pragma unroll 8
    for (int k = 0; k < C_CHUNKS; ++k) {
        float4 v = partial[(size_t)k * B4 + c];
        s0 += v.x; s1 += v.y; s2 += v.z; s3 += v.w;
    }
    out[c] = make_float4(s0, s1, s2, s3);
}

extern "C" void kernel_launch(void* const* d_in, const int* in_sizes, int n_in,
                              void* d_out, int out_size, void* d_ws, size_t ws_size,
                              hipStream_t stream) {
    (void)in_sizes; (void)n_in; (void)out_size; (void)ws_size;

    const float4* x = (const float4*)d_in[0];     // [L, B] fp32, row-major

    // Workspace layout (16B-aligned slabs): cmax | smax | partial, 1 MiB each.
    char* ws = (char*)d_ws;
    float4* cmax    = (float4*)(ws);
    float4* smax    = (float4*)(ws + (size_t)C_CHUNKS * B4 * sizeof(float4));
    float4* partial = (float4*)(ws + (size_t)2 * C_CHUNKS * B4 * sizeof(float4));
    float4* out     = (float4*)d_out;             // B floats = B4 float4

    const int threads = 256;
    dim3 big_grid(B4 / threads, C_CHUNKS);        // 4 x 64 blocks, 2048 waves
    dim3 small_grid(B4 / threads);                // 4 blocks

    sl_chunkmax_kernel <<<big_grid,   threads, 0, stream>>>(x, cmax);
    sl_suffixmax_kernel<<<small_grid, threads, 0, stream>>>(cmax, smax);
    sl_scan_kernel     <<<big_grid,   threads, 0, stream>>>(x, cmax, smax, partial);
    sl_reduce_kernel   <<<small_grid, threads, 0, stream>>>(partial, out);
}